// GATBlock_25907242729823
// MI455X (gfx1250) — compile-verified
//
#include <hip/hip_runtime.h>
#include <hip/hip_bf16.h>

// ---------------------------------------------------------------------------
// GAT block for MI455X (gfx1250, wave32).
// h = x @ W done with V_WMMA_F32_16X16X4_F32 (full-precision f32 WMMA).
// Edge softmax + scatter are streaming/atomic kernels (bandwidth bound).
// ---------------------------------------------------------------------------

typedef float v2f __attribute__((ext_vector_type(2)));
typedef float v8f __attribute__((ext_vector_type(8)));

#define HEADS 4
#define HDIM  16
#define FEAT  64
#define NEG_SLOPE 0.2f
#define BN_EPS 1e-5f

// -------------------- float atomic max (sign-split trick) -------------------
__device__ __forceinline__ void atomicMaxFloat(float* addr, float val) {
    if (!__signbitf(val)) {
        atomicMax(reinterpret_cast<int*>(addr), __float_as_int(val));
    } else {
        atomicMin(reinterpret_cast<unsigned int*>(addr), __float_as_uint(val));
    }
}

// -------------------- kernel 1: init workspace ------------------------------
__global__ void init_k(const float* __restrict__ prev, const float* __restrict__ bias,
                       float* __restrict__ y, float* __restrict__ emax,
                       float* __restrict__ denom, float* __restrict__ colsum,
                       float* __restrict__ colsumsq, int nNodes) {
    int i = blockIdx.x * blockDim.x + threadIdx.x;
    int totalY = nNodes * FEAT;
    if (i < totalY) y[i] = prev[i] + bias[i & (FEAT - 1)];
    if (i < nNodes * HEADS) { emax[i] = -__builtin_inff(); denom[i] = 0.0f; }
    if (i < FEAT) { colsum[i] = 0.0f; colsumsq[i] = 0.0f; }
}

// -------------------- kernel 2: h = x @ W via WMMA f32 16x16x4 --------------
// One wave per 16x16 output tile. K = 64 -> 16 chained WMMAs.
// A 16x4 layout: lanes 0-15 M=0..15 {V0:K=0, V1:K=1}; lanes 16-31 {V0:K=2, V1:K=3}
// B 4x16 layout: lanes hold N; lanes 0-15 {V0:K=0, V1:K=1}; lanes 16-31 {V0:K=2, V1:K=3}
// C 16x16:       VGPR v: M = v + 8*(lane>=16), N = lane&15
__global__ void gemm_wmma_k(const float* __restrict__ x, const float* __restrict__ W,
                            float* __restrict__ h, int nNodes, int totalTiles) {
    int wave = threadIdx.x >> 5;                       // 4 waves per block of 128
    int tile = blockIdx.x * 4 + wave;                  // wave-uniform
    if (tile >= totalTiles) return;                    // whole-wave exit: EXEC stays all-1
    int mt = tile >> 2;                                // 4 N-tiles (FEAT/16)
    int nt = tile & 3;

    int lane  = threadIdx.x & 31;
    int r     = lane & 15;
    int halfK = lane >> 4;                             // 0 or 1

    int row = mt * 16 + r;
    if (row >= nNodes) row = nNodes - 1;               // defensive (N divisible by 16 anyway)
    int col = nt * 16 + r;

    v8f c = {};
    #pragma unroll
    for (int k0 = 0; k0 < FEAT; k0 += 4) {
        int ka = k0 + 2 * halfK;
        v2f a, b;
        a.x = x[row * FEAT + ka + 0];
        a.y = x[row * FEAT + ka + 1];
        b.x = W[(ka + 0) * FEAT + col];
        b.y = W[(ka + 1) * FEAT + col];
        c = __builtin_amdgcn_wmma_f32_16x16x4_f32(
                /*neg_a=*/false, a, /*neg_b=*/false, b,
                /*c_mod=*/(short)0, c, /*reuse_a=*/false, /*reuse_b=*/false);
    }

    #pragma unroll
    for (int v = 0; v < 8; ++v) {
        int m = mt * 16 + v + 8 * halfK;
        if (m < nNodes) h[m * FEAT + nt * 16 + r] = c[v];
    }
}

// -------------------- kernel 3: attention coefficients ----------------------
__global__ void attn_coef_k(const float* __restrict__ h, const float* __restrict__ att_src,
                            const float* __restrict__ att_dst, float* __restrict__ a_s,
                            float* __restrict__ a_d, int nNodes) {
    int n = blockIdx.x * blockDim.x + threadIdx.x;
    if (n >= nNodes) return;
    #pragma unroll
    for (int hd = 0; hd < HEADS; ++hd) {
        float ss = 0.0f, sd = 0.0f;
        #pragma unroll
        for (int cidx = 0; cidx < HDIM; ++cidx) {
            float v = h[n * FEAT + hd * HDIM + cidx];
            ss += v * att_src[hd * HDIM + cidx];
            sd += v * att_dst[hd * HDIM + cidx];
        }
        a_s[n * HEADS + hd] = ss;
        a_d[n * HEADS + hd] = sd;
    }
}

// -------------------- edge helpers ------------------------------------------
__device__ __forceinline__ void edge_endpoints(const int* __restrict__ ei, int e,
                                               int E, int& s, int& d) {
    if (e < E) { s = ei[e]; d = ei[E + e]; }
    else       { s = e - E; d = s; }                   // self loop
}

__device__ __forceinline__ float lrelu(float v) {
    return v > 0.0f ? v : NEG_SLOPE * v;
}

// -------------------- kernel 4: segment max over dst ------------------------
__global__ void edge_max_k(const int* __restrict__ ei, const float* __restrict__ a_s,
                           const float* __restrict__ a_d, float* __restrict__ emax,
                           int E, int nNodes) {
    int e = blockIdx.x * blockDim.x + threadIdx.x;
    if (e >= E + nNodes) return;
    int s, d; edge_endpoints(ei, e, E, s, d);
    float4 vs = *reinterpret_cast<const float4*>(a_s + 4 * s);
    float4 vd = *reinterpret_cast<const float4*>(a_d + 4 * d);
    atomicMaxFloat(emax + d * 4 + 0, lrelu(vs.x + vd.x));
    atomicMaxFloat(emax + d * 4 + 1, lrelu(vs.y + vd.y));
    atomicMaxFloat(emax + d * 4 + 2, lrelu(vs.z + vd.z));
    atomicMaxFloat(emax + d * 4 + 3, lrelu(vs.w + vd.w));
}

// -------------------- kernel 5: segment sum of exp --------------------------
__global__ void edge_den_k(const int* __restrict__ ei, const float* __restrict__ a_s,
                           const float* __restrict__ a_d, const float* __restrict__ emax,
                           float* __restrict__ denom, int E, int nNodes) {
    int e = blockIdx.x * blockDim.x + threadIdx.x;
    if (e >= E + nNodes) return;
    int s, d; edge_endpoints(ei, e, E, s, d);
    float4 vs = *reinterpret_cast<const float4*>(a_s + 4 * s);
    float4 vd = *reinterpret_cast<const float4*>(a_d + 4 * d);
    float4 mx = *reinterpret_cast<const float4*>(emax + 4 * d);
    atomicAdd(denom + d * 4 + 0, __expf(lrelu(vs.x + vd.x) - mx.x));
    atomicAdd(denom + d * 4 + 1, __expf(lrelu(vs.y + vd.y) - mx.y));
    atomicAdd(denom + d * 4 + 2, __expf(lrelu(vs.z + vd.z) - mx.z));
    atomicAdd(denom + d * 4 + 3, __expf(lrelu(vs.w + vd.w) - mx.w));
}

// -------------------- kernel 6: normalized scatter of messages --------------
// one thread per (edge, head); 16 float atomic adds each
__global__ void edge_scatter_k(const int* __restrict__ ei, const float* __restrict__ a_s,
                               const float* __restrict__ a_d, const float* __restrict__ emax,
                               const float* __restrict__ denom, const float* __restrict__ h,
                               float* __restrict__ y, int E, int nNodes) {
    int i = blockIdx.x * blockDim.x + threadIdx.x;
    if (i >= (E + nNodes) * HEADS) return;
    int e  = i >> 2;
    int hd = i & 3;
    int s, d; edge_endpoints(ei, e, E, s, d);
    float t = lrelu(a_s[s * 4 + hd] + a_d[d * 4 + hd]);
    float alpha = __expf(t - emax[d * 4 + hd]) / denom[d * 4 + hd];
    const float* hs = h + s * FEAT + hd * HDIM;
    float*       yd = y + d * FEAT + hd * HDIM;
    #pragma unroll
    for (int cidx = 0; cidx < HDIM; ++cidx)
        atomicAdd(yd + cidx, hs[cidx] * alpha);
}

// -------------------- kernel 7: BN column reduction -------------------------
// block = 256 threads = 64 features x 4 row-groups; 64 nodes per block
__global__ void bn_reduce_k(const float* __restrict__ y, float* __restrict__ colsum,
                            float* __restrict__ colsumsq, int nNodes) {
    __shared__ float s1[256];
    __shared__ float s2[256];
    int f  = threadIdx.x & 63;
    int r0 = threadIdx.x >> 6;                 // 0..3
    int base = blockIdx.x * 64;
    float s = 0.0f, q = 0.0f;
    for (int r = r0; r < 64; r += 4) {
        int n = base + r;
        if (n < nNodes) {
            float v = y[n * FEAT + f];
            s += v; q += v * v;
        }
    }
    s1[threadIdx.x] = s;
    s2[threadIdx.x] = q;
    __syncthreads();
    if (r0 == 0) {
        s = s1[f] + s1[f + 64] + s1[f + 128] + s1[f + 192];
        q = s2[f] + s2[f + 64] + s2[f + 128] + s2[f + 192];
        atomicAdd(colsum + f, s);
        atomicAdd(colsumsq + f, q);
    }
}

// -------------------- kernel 8: BN apply + ReLU -----------------------------
__global__ void bn_apply_k(const float* __restrict__ y, const float* __restrict__ colsum,
                           const float* __restrict__ colsumsq, const float* __restrict__ gamma,
                           const float* __restrict__ beta, float* __restrict__ out, int nNodes) {
    int i = blockIdx.x * blockDim.x + threadIdx.x;
    if (i >= nNodes * FEAT) return;
    int f = i & (FEAT - 1);
    float invN = 1.0f / (float)nNodes;
    float mean = colsum[f] * invN;
    float var  = colsumsq[f] * invN - mean * mean;
    float rs   = rsqrtf(var + BN_EPS);
    float v    = (y[i] - mean) * rs * gamma[f] + beta[f];
    out[i] = v > 0.0f ? v : 0.0f;
}

// ---------------------------------------------------------------------------
extern "C" void kernel_launch(void* const* d_in, const int* in_sizes, int n_in,
                              void* d_out, int out_size, void* d_ws, size_t ws_size,
                              hipStream_t stream) {
    const float* prev    = (const float*)d_in[0];
    const float* x       = (const float*)d_in[1];
    const int*   ei      = (const int*)d_in[2];
    const float* W       = (const float*)d_in[3];
    const float* att_src = (const float*)d_in[4];
    const float* att_dst = (const float*)d_in[5];
    const float* bias    = (const float*)d_in[6];
    const float* gamma   = (const float*)d_in[7];
    const float* beta    = (const float*)d_in[8];
    float*       out     = (float*)d_out;

    const int N = in_sizes[0] / FEAT;     // 50000
    const int E = in_sizes[2] / 2;        // 800000
    const int ET = E + N;                 // with self loops

    // workspace layout (floats)
    float* ws    = (float*)d_ws;
    float* h     = ws;                                   // N*64
    float* y     = h     + (size_t)N * FEAT;             // N*64
    float* a_s   = y     + (size_t)N * FEAT;             // N*4
    float* a_d   = a_s   + (size_t)N * HEADS;            // N*4
    float* emax  = a_d   + (size_t)N * HEADS;            // N*4
    float* denom = emax  + (size_t)N * HEADS;            // N*4
    float* csum  = denom + (size_t)N * HEADS;            // 64
    float* csq   = csum  + FEAT;                         // 64

    const int B = 256;

    // 1) init
    init_k<<<(N * FEAT + B - 1) / B, B, 0, stream>>>(prev, bias, y, emax, denom, csum, csq, N);

    // 2) h = x @ W  (WMMA)
    int mtiles = (N + 15) / 16;
    int totalTiles = mtiles * (FEAT / 16);
    gemm_wmma_k<<<(totalTiles + 3) / 4, 128, 0, stream>>>(x, W, h, N, totalTiles);

    // 3) attention coefficients
    attn_coef_k<<<(N + B - 1) / B, B, 0, stream>>>(h, att_src, att_dst, a_s, a_d, N);

    // 4) segment max
    edge_max_k<<<(ET + B - 1) / B, B, 0, stream>>>(ei, a_s, a_d, emax, E, N);

    // 5) segment sum of exp
    edge_den_k<<<(ET + B - 1) / B, B, 0, stream>>>(ei, a_s, a_d, emax, denom, E, N);

    // 6) scatter alpha * h[src] into y[dst]
    edge_scatter_k<<<(ET * HEADS + B - 1) / B, B, 0, stream>>>(ei, a_s, a_d, emax, denom, h, y, E, N);

    // 7) BN column stats
    bn_reduce_k<<<(N + 63) / 64, 256, 0, stream>>>(y, csum, csq, N);

    // 8) BN apply + ReLU
    bn_apply_k<<<(N * FEAT + B - 1) / B, B, 0, stream>>>(y, csum, csq, gamma, beta, out, N);
}